// GNNEncoder_3736621548265
// MI455X (gfx1250) — compile-verified
//
#include <hip/hip_runtime.h>

typedef float v2f __attribute__((ext_vector_type(2)));
typedef float v8f __attribute__((ext_vector_type(8)));

#define HID 128
#define LN_EPS 1e-5f

// ---------------------------------------------------------------------------
// Fused GEMM (A[N,K] @ W[K,128] + bias) -> SiLU -> LayerNorm(gamma,beta)
// One wave computes a 16-row strip x 128 cols using V_WMMA_F32_16X16X4_F32.
// ---------------------------------------------------------------------------
template <int K>
__global__ __launch_bounds__(256) void gemm_silu_ln_kernel(
    const float* __restrict__ A, const float* __restrict__ W,
    const float* __restrict__ bias, const float* __restrict__ gamma,
    const float* __restrict__ beta, float* __restrict__ out, int nRows) {
  // W staged transposed into LDS: element (k, n) stored at
  //   sW[n*K + (k ^ (2*(n&15)))]   (XOR swizzle keeps float2 pairs intact and
  // spreads the 16 lanes of a half-wave over 16 distinct banks; no padding).
  __shared__ float sW[HID * K];
  for (int idx = threadIdx.x; idx < K * HID; idx += 256) {
    int k = idx >> 7;          // idx / 128  (row of W)
    int n = idx & (HID - 1);   // idx % 128  (col of W)
    sW[n * K + (k ^ ((n & 15) << 1))] = W[idx];
  }
  __syncthreads();

  const int lane = threadIdx.x & 31;
  const int wave = threadIdx.x >> 5;
  const int l16  = lane & 15;
  const int hi   = lane >> 4;  // 0: lanes 0-15 (K pair {k0,k0+1}), 1: {k0+2,k0+3}
  const int row0 = (blockIdx.x * 8 + wave) * 16;
  if (row0 >= nRows) return;   // wave-uniform: WMMA below runs with EXEC all-1s

  const float* arow = A + (size_t)(row0 + l16) * K + 2 * hi;

  v8f acc[8];
#pragma unroll
  for (int t = 0; t < 8; ++t)
    acc[t] = (v8f){0.f, 0.f, 0.f, 0.f, 0.f, 0.f, 0.f, 0.f};

#pragma unroll 4
  for (int k0 = 0; k0 < K; k0 += 4) {
    // A fragment: row = row0 + l16, K elems {k0+2*hi, k0+2*hi+1}
    v2f av = *(const v2f*)(arow + k0);
#pragma unroll
    for (int t = 0; t < 8; ++t) {
      const int col = t * 16 + l16;
      // B fragment: W[k0+2*hi .. +1][col] from swizzled transposed LDS image
      v2f bv = *(const v2f*)(&sW[col * K + ((k0 + 2 * hi) ^ (l16 << 1))]);
      acc[t] = __builtin_amdgcn_wmma_f32_16x16x4_f32(
          false, av, false, bv, (short)0, acc[t], false, false);
    }
  }

  // bias + SiLU, accumulate row statistics. C/D layout: VGPR j, this lane's
  // half holds element (M = j + 8*hi, N = t*16 + l16).
  float s[8], q[8];
#pragma unroll
  for (int j = 0; j < 8; ++j) { s[j] = 0.f; q[j] = 0.f; }
#pragma unroll
  for (int t = 0; t < 8; ++t) {
    const float bb = bias[t * 16 + l16];
#pragma unroll
    for (int j = 0; j < 8; ++j) {
      float p = acc[t][j] + bb;
      float u = p / (1.f + __expf(-p));  // SiLU
      acc[t][j] = u;
      s[j] += u;
      q[j] += u * u;
    }
  }
  // Reduce across the 16 lanes of each half (masks 1..8 never cross bit 4).
#pragma unroll
  for (int j = 0; j < 8; ++j) {
#pragma unroll
    for (int m = 1; m <= 8; m <<= 1) {
      s[j] += __shfl_xor(s[j], m, 32);
      q[j] += __shfl_xor(q[j], m, 32);
    }
  }
  float mean[8], rstd[8];
  const float inv = 1.f / (float)HID;
#pragma unroll
  for (int j = 0; j < 8; ++j) {
    mean[j] = s[j] * inv;
    float var = q[j] * inv - mean[j] * mean[j];
    rstd[j] = rsqrtf(var + LN_EPS);
  }
#pragma unroll
  for (int t = 0; t < 8; ++t) {
    const int col = t * 16 + l16;
    const float gm = gamma[col];
    const float bt = beta[col];
    float* orow = out + (size_t)(row0 + 8 * hi) * HID + col;
#pragma unroll
    for (int j = 0; j < 8; ++j)
      orow[(size_t)j * HID] = (acc[t][j] - mean[j]) * rstd[j] * gm + bt;
  }
}

// ---------------------------------------------------------------------------
__global__ __launch_bounds__(256) void zero_kernel(float* __restrict__ p,
                                                   size_t n) {
  size_t i = (size_t)blockIdx.x * 256 + threadIdx.x;
  if (i < n) p[i] = 0.f;
}

__global__ __launch_bounds__(256) void deg_kernel(const int* __restrict__ dst,
                                                  float* __restrict__ deg,
                                                  int nEdges) {
  int e = blockIdx.x * 256 + threadIdx.x;
  if (e < nEdges) atomicAdd(deg + dst[e], 1.f);
}

__global__ __launch_bounds__(256) void invdeg_kernel(float* __restrict__ deg,
                                                     int n) {
  int i = blockIdx.x * 256 + threadIdx.x;
  if (i < n) {
    float d = deg[i];
    deg[i] = d > 0.f ? 1.f / d : 0.f;
  }
}

// agg[dst[e], :] += z[src[e], :]   (flat over edge x col; src/dst wave-uniform)
__global__ __launch_bounds__(256) void scatter_add_kernel(
    const float* __restrict__ z, const int* __restrict__ src,
    const int* __restrict__ dst, float* __restrict__ agg, int nEdges) {
  int tid = blockIdx.x * 256 + threadIdx.x;
  int e = tid >> 7;
  if (e >= nEdges) return;
  int c = tid & (HID - 1);
  int si = src[e];
  int di = dst[e];
  atomicAdd(agg + (size_t)di * HID + c, z[(size_t)si * HID + c]);
}

// h += agg * inv_deg  (residual + mean aggregation)
__global__ __launch_bounds__(256) void update_kernel(
    float* __restrict__ h, const float* __restrict__ agg,
    const float* __restrict__ invdeg, int nNodes) {
  int i = blockIdx.x * 256 + threadIdx.x;
  if (i < nNodes * HID) h[i] += agg[i] * invdeg[i >> 7];
}

// ---------------------------------------------------------------------------
extern "C" void kernel_launch(void* const* d_in, const int* in_sizes, int n_in,
                              void* d_out, int out_size, void* d_ws,
                              size_t ws_size, hipStream_t stream) {
  const float* x   = (const float*)d_in[0];
  const int*   ei  = (const int*)d_in[1];   // [2, E] flat
  const float* W0  = (const float*)d_in[2]; // [32, 128]
  const float* b0  = (const float*)d_in[3];
  const float* g0  = (const float*)d_in[4];
  const float* be0 = (const float*)d_in[5];
  const float* Wls = (const float*)d_in[6]; // [L, 128, 128]
  const float* bls = (const float*)d_in[7];
  const float* gls = (const float*)d_in[8];
  const float* bes = (const float*)d_in[9];
  float* h = (float*)d_out;                 // [N, 128], final output

  const int N = in_sizes[0] / 32;
  const int E = in_sizes[1] / 2;
  const int L = in_sizes[6] / (HID * HID);
  const int* srcIdx = ei;
  const int* dstIdx = ei + E;

  // Workspace layout: inv_deg | z | agg
  char* ws = (char*)d_ws;
  float* invdeg = (float*)ws;
  size_t off = (((size_t)N * sizeof(float)) + 255) & ~(size_t)255;
  float* z = (float*)(ws + off);
  off += (((size_t)N * HID * sizeof(float)) + 255) & ~(size_t)255;
  float* agg = (float*)(ws + off);

  const size_t NH = (size_t)N * HID;
  const unsigned nhBlocks   = (unsigned)((NH + 255) / 256);
  const unsigned nBlocks    = (unsigned)((N + 255) / 256);
  const unsigned eBlocks    = (unsigned)((E + 255) / 256);
  const unsigned ecBlocks   = (unsigned)(((size_t)E * HID + 255) / 256);
  const unsigned gemmBlocks = (unsigned)((N + 127) / 128);  // 8 strips/block

  // in-degree -> inverse degree (in place)
  zero_kernel<<<nBlocks, 256, 0, stream>>>(invdeg, (size_t)N);
  deg_kernel<<<eBlocks, 256, 0, stream>>>(dstIdx, invdeg, E);
  invdeg_kernel<<<nBlocks, 256, 0, stream>>>(invdeg, N);

  // h = LN(SiLU(x @ W0 + b0))
  gemm_silu_ln_kernel<32><<<gemmBlocks, 256, 0, stream>>>(
      x, W0, b0, g0, be0, h, N);

  for (int i = 0; i < L; ++i) {
    zero_kernel<<<nhBlocks, 256, 0, stream>>>(agg, NH);
    gemm_silu_ln_kernel<HID><<<gemmBlocks, 256, 0, stream>>>(
        h, Wls + (size_t)i * HID * HID, bls + (size_t)i * HID,
        gls + (size_t)i * HID, bes + (size_t)i * HID, z, N);
    scatter_add_kernel<<<ecBlocks, 256, 0, stream>>>(z, srcIdx, dstIdx, agg, E);
    update_kernel<<<nhBlocks, 256, 0, stream>>>(h, agg, invdeg, N);
  }
}